// LocalNeighborhood_2482491097340
// MI455X (gfx1250) — compile-verified
//
#include <hip/hip_runtime.h>

// Problem constants from the reference
#define BB 8
#define LL 4096
#define KK 16
#define DD 128

typedef __attribute__((ext_vector_type(2))) float v2f;
typedef __attribute__((ext_vector_type(4))) float v4f;
typedef __attribute__((ext_vector_type(8))) float v8f;

// ---------------------------------------------------------------------------
// Kernel 0: per-point feature = (x, y, z, |c|^2) from frame[:,:,0,:]
// frame layout (B,L,4,3): point i=(b*L+l) center at flat offset i*12.
// ---------------------------------------------------------------------------
__global__ void ln_feat_kernel(const float* __restrict__ frame,
                               v4f* __restrict__ feat) {
  int i = blockIdx.x * blockDim.x + threadIdx.x;
  if (i >= BB * LL) return;
  const float* c = frame + (size_t)i * 12;
  float x = c[0], y = c[1], z = c[2];
  v4f f; f.x = x; f.y = y; f.z = z; f.w = x * x + y * y + z * z;
  feat[i] = f;
}

// ---------------------------------------------------------------------------
// Main kernel: one wave32 per 16-query tile.
//   distance proxy d[m][n] = |c_m|^2 - 2*c_m.c_n   (row-constant |c_n|^2 drop)
//   A (16x4, M=candidates): lanes 0-15 hold (x,y) = K0,K1; lanes 16-31 (z,|c|^2)=K2,K3
//   B (4x16, N=queries):    lanes 0-15 hold (-2x,-2y); lanes 16-31 (-2z, 1)
//   C (16x16): lane l<16 gets D[v][l] v=0..7; lane l>=16 gets D[v+8][l-16]
//   => lane owns 8 candidate distances for ONE query -> register top-16.
// ---------------------------------------------------------------------------
__global__ __launch_bounds__(128) void ln_knn_kernel(
    const float* __restrict__ frame,
    const float* __restrict__ attr,
    const v4f* __restrict__ feat,
    float* __restrict__ out_euclid,   // (B,L,16,3)
    float* __restrict__ out_attr)     // (B,L,16,128)
{
  const int lane  = threadIdx.x & 31;
  const int wib   = threadIdx.x >> 5;
  const int tile  = blockIdx.x * (blockDim.x >> 5) + wib;  // 0 .. B*L/16-1
  const int b     = tile >> 8;                              // 256 tiles per batch
  const int q0    = (tile & 255) * 16;                      // query base in batch
  const int half  = lane >> 4;                              // 0 | 1
  const int l16   = lane & 15;

  const v4f* featB = feat + (size_t)b * LL;

  // ---- B matrix (queries): constant across all candidate tiles
  v4f fq = featB[q0 + l16];
  v2f bmat;
  if (half == 0) { bmat.x = -2.0f * fq.x; bmat.y = -2.0f * fq.y; }
  else           { bmat.x = -2.0f * fq.z; bmat.y = 1.0f; }

  // ---- lane-local sorted (ascending) top-16 of this lane's candidate half
  float tv[16]; int ti[16];
#pragma unroll
  for (int k = 0; k < 16; ++k) { tv[k] = __builtin_inff(); ti[k] = 0; }

  const v2f* feat2 = (const v2f*)featB;
  const int  NTILES = LL / 16;

  // software pipeline: A-tile load for ct+1 overlaps WMMA+selection of ct
  v2f fa = feat2[(size_t)l16 * 2 + half];   // tile 0

  for (int ct = 0; ct < NTILES; ++ct) {
    const int m0 = ct * 16;
    v2f amat = fa;
    // issue next tile's A load now; selection work below hides its latency
    {
      int nct = ct + 1;
      if (nct == NTILES) nct = 0;           // harmless dummy on last iter
      fa = feat2[(size_t)(nct * 16 + l16) * 2 + half];
    }

    v8f c = {0.f, 0.f, 0.f, 0.f, 0.f, 0.f, 0.f, 0.f};
    c = __builtin_amdgcn_wmma_f32_16x16x4_f32(false, amat, false, bmat,
                                              (short)0, c, false, false);
    const int mbase = m0 + half * 8;
#pragma unroll
    for (int v = 0; v < 8; ++v) {
      float val = c[v];
      int   idx = mbase + v;
      if (val < tv[15]) {            // quick reject vs current 16th-best
        float cv = val; int ci = idx;
#pragma unroll
        for (int k = 0; k < 16; ++k) {  // bubble-insert, old max falls off
          bool sw  = cv < tv[k];
          float nv = sw ? cv : tv[k];
          int   ni = sw ? ci : ti[k];
          cv = sw ? tv[k] : cv;
          ci = sw ? ti[k] : ci;
          tv[k] = nv; ti[k] = ni;
        }
      }
    }
  }

  // ---- merge lane l with lane l^16 (complementary candidate halves)
  float bv[16]; int bi[16];
#pragma unroll
  for (int k = 0; k < 16; ++k) {
    bv[k] = __shfl_xor(tv[k], 16, 32);
    bi[k] = __shfl_xor(ti[k], 16, 32);
  }
  // min-stage against reversed partner -> bitonic sequence of the lowest 16
  float mv[16]; int mi[16];
#pragma unroll
  for (int k = 0; k < 16; ++k) {
    bool takeA = tv[k] <= bv[15 - k];
    mv[k] = takeA ? tv[k] : bv[15 - k];
    mi[k] = takeA ? ti[k] : bi[15 - k];
  }
  // bitonic merge network (static indices only)
#pragma unroll
  for (int d = 8; d >= 1; d >>= 1) {
#pragma unroll
    for (int k = 0; k < 16; ++k) {
      if ((k & d) == 0) {
        bool sw  = mv[k] > mv[k + d];
        float lv = sw ? mv[k + d] : mv[k];
        int   li = sw ? mi[k + d] : mi[k];
        float hv = sw ? mv[k] : mv[k + d];
        int   hi = sw ? mi[k] : mi[k + d];
        mv[k] = lv; mi[k] = li; mv[k + d] = hv; mi[k + d] = hi;
      }
    }
  }
  // lanes 0-15 now hold the final ascending-distance top-16 for query q0+l16
  // (lanes 16-31 hold an identical copy up to fp ties)

  // ---- euclidian: e[k][r] = sum_c delta[k][c] * frame[b,q,1+c,r]
  if (lane < 16) {
    const int q = q0 + lane;
    const float* fr = frame + ((size_t)(b * LL + q) * 4 + 1) * 3;  // 9 floats
    float r00 = fr[0], r01 = fr[1], r02 = fr[2];
    float r10 = fr[3], r11 = fr[4], r12 = fr[5];
    float r20 = fr[6], r21 = fr[7], r22 = fr[8];
    float cx = fq.x, cy = fq.y, cz = fq.z;
    float* eo = out_euclid + (size_t)(b * LL + q) * (KK * 3);
#pragma unroll
    for (int k = 0; k < 16; ++k) {
      v4f fn = featB[mi[k]];
      float dx = fn.x - cx, dy = fn.y - cy, dz = fn.z - cz;
      // streaming output: non-temporal, keep L2 for attr/feat reuse
      __builtin_nontemporal_store(dx * r00 + dy * r10 + dz * r20, eo + k * 3 + 0);
      __builtin_nontemporal_store(dx * r01 + dy * r11 + dz * r21, eo + k * 3 + 1);
      __builtin_nontemporal_store(dx * r02 + dy * r12 + dz * r22, eo + k * 3 + 2);
    }
  }

  // ---- gathered_attr: full wave copies 512B per (query, neighbor)
  // 32 lanes x b128 == 128 floats == one attr row.
  // Loads of attr are L2-resident (2MB/batch); stores are non-temporal
  // so 268MB of output doesn't thrash the 192MB L2.
  for (int qq = 0; qq < 16; ++qq) {
#pragma unroll
    for (int k = 0; k < 16; ++k) {
      int nb = __shfl(mi[k], qq, 32);   // neighbor k of query q0+qq
      const v4f* src = (const v4f*)(attr + ((size_t)b * LL + nb) * DD);
      v4f* dst = (v4f*)(out_attr +
                 (((size_t)(b * LL + q0 + qq) * KK + k) * DD));
      v4f row = src[lane];
      __builtin_nontemporal_store(row, dst + lane);
    }
  }
}

// ---------------------------------------------------------------------------
extern "C" void kernel_launch(void* const* d_in, const int* in_sizes, int n_in,
                              void* d_out, int out_size, void* d_ws, size_t ws_size,
                              hipStream_t stream) {
  const float* frame = (const float*)d_in[0];   // (B,L,4,3) fp32
  const float* attr  = (const float*)d_in[1];   // (B,L,D)   fp32

  v4f* feat = (v4f*)d_ws;                       // B*L*16B = 512 KB scratch

  float* out_euclid = (float*)d_out;                              // (B,L,16,3)
  float* out_attr   = out_euclid + (size_t)BB * LL * KK * 3;      // (B,L,16,128)

  ln_feat_kernel<<<(BB * LL + 255) / 256, 256, 0, stream>>>(frame, feat);

  // one wave per 16-query tile; 4 waves per block -> 2048 waves total
  const int tiles = BB * (LL / 16);
  ln_knn_kernel<<<tiles / 4, 128, 0, stream>>>(frame, attr, feat,
                                               out_euclid, out_attr);
}